// SVDReduce_76605036691884
// MI455X (gfx1250) — compile-verified
//
#include <hip/hip_runtime.h>
#include <stdint.h>

// out = X[:, :32, :32]  (SVD reconstruction of a full-rank matrix is the identity)
// X:   8192 x 64 x 64 f32   (16384 B per batch)
// out: 8192 x 32 x 32 f32   ( 4096 B per batch)
//
// Pure data movement -> use the Tensor Data Mover (TDM):
//   tensor_load_to_lds : strided 3-D gather (32x32 crop of each 64x64 slab) -> dense LDS
//   tensor_store_from_lds : dense LDS -> contiguous output
// One wave per workgroup issues both ops; TENSORcnt orders them.

typedef unsigned int u32x4 __attribute__((ext_vector_type(4)));
typedef int          i32x4 __attribute__((ext_vector_type(4)));
typedef int          i32x8 __attribute__((ext_vector_type(8)));

#define TB 8  // batches per workgroup: LDS tile = TB*32*32*4 = 32 KiB

__global__ __launch_bounds__(32) void svd_crop_tdm_kernel(const float* __restrict__ X,
                                                          float* __restrict__ out) {
    extern __shared__ char lds_buf[];  // dynamic LDS, starts at LDS offset 0
    (void)lds_buf;

    const uint32_t blk = blockIdx.x;
    const uint64_t in_addr  = (uint64_t)(uintptr_t)X   + (uint64_t)blk * (TB * 64ull * 64ull * 4ull);
    const uint64_t out_addr = (uint64_t)(uintptr_t)out + (uint64_t)blk * (TB * 32ull * 32ull * 4ull);

    // ---------------- Load descriptor: 3-D tile [z=TB][y=32][x=32] of f32 ----------------
    // tensor_dim0=64, tensor_dim1=64, tensor_dim2=8192; stride0=64, stride1=4096 (elements)
    u32x4 l_g0;
    l_g0.x = 1u;                                              // count=1 (valid), no gather
    l_g0.y = 0u;                                              // lds_addr = 0
    l_g0.z = (uint32_t)(in_addr & 0xFFFFFFFFu);               // global_addr[31:0]
    l_g0.w = (uint32_t)((in_addr >> 32) & 0x1FFFFFFu)         // global_addr[56:32]
             | (2u << 30);                                    // type = 2 ("image")

    i32x8 l_g1;
    l_g1[0] = (int)(2u << 16);                 // workgroup_mask=0, data_size=2 (4 bytes)
    l_g1[1] = (int)(64u << 16);                // atomic_barrier_addr=0 | tensor_dim0[15:0]=64
    l_g1[2] = (int)(64u << 16);                // tensor_dim0[31:16]=0 | tensor_dim1[15:0]=64
    l_g1[3] = (int)(32u << 16);                // tensor_dim1[31:16]=0 | tile_dim0=32
    l_g1[4] = (int)(32u | ((unsigned)TB << 16)); // tile_dim1=32 | tile_dim2=TB
    l_g1[5] = 64;                              // tensor_dim0_stride[31:0] = 64
    l_g1[6] = (int)(4096u << 16);              // stride0[47:32]=0 | tensor_dim1_stride[15:0]=4096
    l_g1[7] = 0;                               // tensor_dim1_stride[47:16] = 0

    i32x4 l_g2;
    l_g2[0] = 8192;                            // tensor_dim2 = 8192
    l_g2[1] = 0;                               // tensor_dim3 (unused)
    l_g2[2] = 0;                               // tensor_dim2_stride[31:0] (unused, 3-D tile)
    l_g2[3] = 0;                               // stride2 hi | tile_dim3=0 (unused)

    i32x4 l_g3 = {0, 0, 0, 0};                 // dim3 stride / dim4 / tile_dim4 unused

    // ---------------- Store descriptor: 2-D dense tile [y=TB][x=1024] of f32 ----------------
    u32x4 s_g0;
    s_g0.x = 1u;
    s_g0.y = 0u;                                              // lds_addr = 0
    s_g0.z = (uint32_t)(out_addr & 0xFFFFFFFFu);
    s_g0.w = (uint32_t)((out_addr >> 32) & 0x1FFFFFFu) | (2u << 30);

    i32x8 s_g1;
    s_g1[0] = (int)(2u << 16);                 // data_size = 4 bytes
    s_g1[1] = (int)(1024u << 16);              // tensor_dim0[15:0] = 1024
    s_g1[2] = (int)(8192u << 16);              // dim0 hi=0 | tensor_dim1[15:0] = 8192
    s_g1[3] = (int)(1024u << 16);              // dim1 hi=0 | tile_dim0 = 1024
    s_g1[4] = TB;                              // tile_dim1 = TB | tile_dim2 = 0 (2-D)
    s_g1[5] = 1024;                            // tensor_dim0_stride = 1024 (dense)
    s_g1[6] = 0;
    s_g1[7] = 0;

    i32x4 s_g2 = {0, 0, 0, 0};
    i32x4 s_g3 = {0, 0, 0, 0};

#if __clang_major__ >= 23
    i32x8 zero8 = {0, 0, 0, 0, 0, 0, 0, 0};
    __builtin_amdgcn_tensor_load_to_lds(l_g0, l_g1, l_g2, l_g3, zero8, 0);
    __builtin_amdgcn_s_wait_tensorcnt(0);      // LDS tile complete before store reads it
    __builtin_amdgcn_tensor_store_from_lds(s_g0, s_g1, s_g2, s_g3, zero8, 0);
#else
    __builtin_amdgcn_tensor_load_to_lds(l_g0, l_g1, l_g2, l_g3, 0);
    __builtin_amdgcn_s_wait_tensorcnt(0);      // LDS tile complete before store reads it
    __builtin_amdgcn_tensor_store_from_lds(s_g0, s_g1, s_g2, s_g3, 0);
#endif
    __builtin_amdgcn_s_wait_tensorcnt(0);      // drain before wave exit (endpgm also waits)
}

extern "C" void kernel_launch(void* const* d_in, const int* in_sizes, int n_in,
                              void* d_out, int out_size, void* d_ws, size_t ws_size,
                              hipStream_t stream) {
    const float* X = (const float*)d_in[0];
    float* out = (float*)d_out;

    const int batches = 8192;
    dim3 grid(batches / TB);   // 1024 workgroups
    dim3 block(32);            // one wave32 each
    size_t lds_bytes = (size_t)TB * 32 * 32 * sizeof(float);  // 32 KiB dynamic LDS

    hipLaunchKernelGGL(svd_crop_tdm_kernel, grid, block, lds_bytes, stream, X, out);
}